// IrrepsScatter_44212393345454
// MI455X (gfx1250) — compile-verified
//
#include <hip/hip_runtime.h>
#include <hip/hip_bf16.h>
#include <stdint.h>

// Problem constants (match reference)
#define N_NODES   200000
#define DIM       480
#define N_SEG     4096

// Tiling
#define CPG        16                 // columns per group -> 30 groups, 64B/node panel
#define N_CGROUPS  (DIM / CPG)        // 30
#define TILE_NODES 256                // nodes per staged tile (16 KB per buffer)
#define NPART      8                  // node partitions -> 240 workgroups total
#define TPB        1024               // 32 waves
#define N_TILES    ((N_NODES + TILE_NODES - 1) / TILE_NODES)   // 782

#define ACC_FLOATS   (N_SEG * CPG)                // 65536 floats = 256 KB
#define STAGE_FLOATS (TILE_NODES * CPG)           // 4096 floats  = 16 KB
// acc + 2 data stage buffers + 2 label slabs
#define SMEM_BYTES   ((ACC_FLOATS + 2 * STAGE_FLOATS) * 4 + 2 * TILE_NODES * 4) // 296960 B

// ---------------------------------------------------------------------------
// Kernel 0: zero the output sums (float4) and the counts scratch
// ---------------------------------------------------------------------------
__global__ void seg_zero_kernel(float4* __restrict__ out4, float* __restrict__ counts) {
    int i = blockIdx.x * blockDim.x + threadIdx.x;
    if (i < (N_SEG * DIM) / 4) out4[i] = make_float4(0.f, 0.f, 0.f, 0.f);
    if (i < N_SEG)             counts[i] = 0.0f;
}

// ---------------------------------------------------------------------------
// Kernel 1: label histogram (float counts, exact for small integers)
// ---------------------------------------------------------------------------
__global__ void seg_count_kernel(const int* __restrict__ labels,
                                 float* __restrict__ counts) {
    int i = blockIdx.x * blockDim.x + threadIdx.x;
    if (i < N_NODES) {
        unsafeAtomicAdd(&counts[labels[i]], 1.0f);
    }
}

// ---------------------------------------------------------------------------
// Kernel 2: main segment-sum.
//  blockIdx.x = column group (0..29), blockIdx.y = node partition (0..NPART-1)
//  LDS: [acc 4096x16 f32][stage 2x 256x16 f32][slab 2x 256 i32]
//  Data tiles streamed with double-buffered async global->LDS b128 loads;
//  labels staged with regular loads whose latency overlaps the async DMA.
// ---------------------------------------------------------------------------
__global__ void __launch_bounds__(TPB, 1)
seg_accum_kernel(const float* __restrict__ x,
                 const int*   __restrict__ labels,
                 float*       __restrict__ outSums) {
    extern __shared__ float smem[];
    float* acc   = smem;                               // ACC_FLOATS
    float* stage = smem + ACC_FLOATS;                  // 2 * STAGE_FLOATS
    int*   slab  = (int*)(smem + ACC_FLOATS + 2 * STAGE_FLOATS);  // 2 * TILE_NODES

    const int t    = threadIdx.x;
    const int cg   = blockIdx.x;
    const int part = blockIdx.y;
    const int c0   = cg * CPG;

    // zero private accumulator
    for (int i = t; i < ACC_FLOATS; i += TPB) acc[i] = 0.0f;
    __syncthreads();

    // Each thread loads one 16B chunk of the tile panel:
    //   node_sub = t/4 (0..255), chunk = t%4 (which 16B of the 64B row slice)
    const int node_sub = t >> 2;
    const int chunk    = t & 3;
    const unsigned long long xbase = (unsigned long long)x;

    // async issue of one data tile into buffer buf (0/1); EXEC stays full (clamped)
    auto issue_tile = [&](int tile, int buf) {
        long long n = (long long)tile * TILE_NODES + node_sub;
        if (n > (long long)(N_NODES - 1)) n = N_NODES - 1;   // clamp, keep EXEC full
        unsigned goff = (unsigned)(n * (DIM * 4)) + (unsigned)(c0 * 4) + (unsigned)(chunk * 16);
        unsigned lds  = (unsigned)(uintptr_t)(&stage[buf * STAGE_FLOATS]) + (unsigned)(t * 16);
        asm volatile("global_load_async_to_lds_b128 %0, %1, %2"
                     :: "v"(lds), "v"(goff), "s"(xbase) : "memory");
    };

    // stage this tile's 256 labels into LDS (threads 0..63, one int4 each)
    auto stage_labels = [&](int tile, int buf) {
        if (t < TILE_NODES / 4) {
            long long n4 = (long long)tile * TILE_NODES + (long long)t * 4;
            if (n4 > (long long)(N_NODES - 4)) n4 = N_NODES - 4;  // aligned clamp
            const int4 v = *(const int4*)(labels + n4);
            *((int4*)(slab + buf * TILE_NODES) + t) = v;
        }
    };

    // prologue: prefetch first tile of this partition
    if (part < N_TILES) {
        issue_tile(part, 0);
        stage_labels(part, 0);
    }

    int buf = 0;
    for (int tile = part; tile < N_TILES; tile += NPART) {
        const int nextTile = tile + NPART;
        if (nextTile < N_TILES) {
            issue_tile(nextTile, buf ^ 1);
            stage_labels(nextTile, buf ^ 1);
            // outstanding: current tile + prefetch -> wait until only prefetch left
            asm volatile("s_wait_asynccnt 0x1" ::: "memory");
        } else {
            asm volatile("s_wait_asynccnt 0x0" ::: "memory");
        }
        __syncthreads();   // all waves' loads for `tile` complete -> LDS valid

        const float* sb = &stage[buf * STAGE_FLOATS];
        const int*   lb = slab + buf * TILE_NODES;
        const int n0 = tile * TILE_NODES;
        int nvalid = N_NODES - n0;
        if (nvalid > TILE_NODES) nvalid = TILE_NODES;

        #pragma unroll
        for (int k = 0; k < STAGE_FLOATS / TPB; ++k) {   // 4 elements per thread
            const int e  = t + k * TPB;
            const int nl = e >> 4;                        // node within tile
            if (nl < nvalid) {
                const int col = e & (CPG - 1);
                const int s   = lb[nl];
                unsafeAtomicAdd(&acc[s * CPG + col], sb[e]);   // ds_add_f32
            }
        }
        __syncthreads();   // all reads of buf done before it can be re-filled
        buf ^= 1;
    }

    // flush private accumulator to global sums (atomic across node partitions)
    for (int i = t; i < ACC_FLOATS; i += TPB) {
        const int s = i >> 4;
        const int c = i & (CPG - 1);
        unsafeAtomicAdd(&outSums[s * DIM + c0 + c], acc[i]);
    }
}

// ---------------------------------------------------------------------------
// Kernel 3: divide by max(count, 1) in place (float4; DIM divisible by 4)
// ---------------------------------------------------------------------------
__global__ void seg_mean_kernel(float4* __restrict__ out4,
                                const float* __restrict__ counts) {
    int i = blockIdx.x * blockDim.x + threadIdx.x;
    if (i < (N_SEG * DIM) / 4) {
        const int s = (i * 4) / DIM;
        const float r = 1.0f / fmaxf(counts[s], 1.0f);
        float4 v = out4[i];
        v.x *= r; v.y *= r; v.z *= r; v.w *= r;
        out4[i] = v;
    }
}

// ---------------------------------------------------------------------------
extern "C" void kernel_launch(void* const* d_in, const int* in_sizes, int n_in,
                              void* d_out, int out_size, void* d_ws, size_t ws_size,
                              hipStream_t stream) {
    (void)in_sizes; (void)n_in; (void)out_size; (void)ws_size;

    const float* x      = (const float*)d_in[0];
    const int*   labels = (const int*)d_in[1];
    // d_in[2] = n_segments scalar (fixed at 4096, unused)
    float* out    = (float*)d_out;
    float* counts = (float*)d_ws;   // 4096 floats of scratch

    // 0) zero sums + counts (harness poisons buffers; re-init every call)
    {
        const int n4 = (N_SEG * DIM) / 4;
        seg_zero_kernel<<<(n4 + 255) / 256, 256, 0, stream>>>((float4*)out, counts);
    }
    // 1) counts histogram
    seg_count_kernel<<<(N_NODES + 255) / 256, 256, 0, stream>>>(labels, counts);
    // 2) segment sums (LDS-privatized, async-staged)
    {
        dim3 grid(N_CGROUPS, NPART);
        seg_accum_kernel<<<grid, TPB, SMEM_BYTES, stream>>>(x, labels, out);
    }
    // 3) mean
    {
        const int n4 = (N_SEG * DIM) / 4;
        seg_mean_kernel<<<(n4 + 255) / 256, 256, 0, stream>>>((float4*)out, counts);
    }
}